// Sampler_65283502899224
// MI455X (gfx1250) — compile-verified
//
#include <hip/hip_runtime.h>
#include <hip/hip_bf16.h>
#include <stdint.h>

// ---------------------------------------------------------------------------
// Sampler: out[r] = temp[r]==0 ? argmax(logits[r]) : argmax(logits[r]/temp[r]
//                                                    + gumbel(threefry(key=1)))
// JAX threefry counter pairing: flat element j pairs with j + N/2 ==> row r
// pairs with row r + B/2 at the same column. Each workgroup owns one row pair
// for one column slice, so every threefry2x32 call feeds BOTH rows.
// SPLIT column slices per row pair -> (B/2)*SPLIT WGs for full-chip occupancy;
// partial (val,idx) per chain go to d_ws, finalize kernel reduces them.
// ---------------------------------------------------------------------------

#define WAIT_ASYNC(imm) asm volatile("s_wait_asynccnt " #imm ::: "memory")
#define WAIT_DS0()      asm volatile("s_wait_dscnt 0" ::: "memory")

__device__ __forceinline__ void async_b128(uint32_t lds_off, const float* gptr) {
    // GLOBAL_LOAD_ASYNC_TO_LDS_B128, GV mode, non-temporal (stream-once data).
    asm volatile("global_load_async_to_lds_b128 %0, %1, off th:TH_LOAD_NT"
                 :: "v"(lds_off), "v"((unsigned long long)(uintptr_t)gptr)
                 : "memory");
}

__device__ __forceinline__ uint32_t rotl32(uint32_t x, int d) {
    return (x << d) | (x >> (32 - d));   // -> v_alignbit_b32
}

// Exact JAX threefry2x32 for key = jax.random.key(1) -> (k0,k1) = (0,1).
__device__ __forceinline__ void threefry2x32_k1(uint32_t x0, uint32_t x1,
                                                uint32_t& o0, uint32_t& o1) {
    const uint32_t ks0 = 0u, ks1 = 1u, ks2 = 0x1BD11BDBu; // 0^1^0x1BD11BDA
    x0 += ks0; x1 += ks1;
#define TF_R(r) { x0 += x1; x1 = rotl32(x1, r); x1 ^= x0; }
    TF_R(13) TF_R(15) TF_R(26) TF_R(6)   x0 += ks1; x1 += ks2 + 1u;
    TF_R(17) TF_R(29) TF_R(16) TF_R(24)  x0 += ks2; x1 += ks0 + 2u;
    TF_R(13) TF_R(15) TF_R(26) TF_R(6)   x0 += ks0; x1 += ks1 + 3u;
    TF_R(17) TF_R(29) TF_R(16) TF_R(24)  x0 += ks1; x1 += ks2 + 4u;
    TF_R(13) TF_R(15) TF_R(26) TF_R(6)   x0 += ks2; x1 += ks0 + 5u;
#undef TF_R
    o0 = x0; o1 = x1;
}

__device__ __forceinline__ float gumbel_from_bits(uint32_t bits) {
    const float TINY = 1.175494350822287508e-38f;                // finfo(f32).tiny
    float f = __uint_as_float((bits >> 9) | 0x3f800000u) - 1.0f; // [0,1)
    float u = fmaxf(f, TINY);                                    // jax uniform
    return -__logf(-__logf(u));                                  // 2x v_log_f32
}

#define THREADS 256
#define VEC     4
#define TILE    (THREADS * VEC)   // 1024 floats = 4KB per row per stage
#define STAGES  4                 // async pipeline depth
#define SPLIT   8                 // column slices per row pair

__global__ __launch_bounds__(THREADS)
void sampler_partial_kernel(const float* __restrict__ logits,
                            const float* __restrict__ temps,
                            float* __restrict__ wsV, int* __restrict__ wsI,
                            int B, int V) {
    __shared__ float tiles[STAGES][2][TILE];   // 32 KB async staging
    __shared__ float redV[4][THREADS];
    __shared__ int   redI[4][THREADS];

    const int half = B >> 1;
    const int pair = blockIdx.y;          // rows (pair, pair+half)
    const int spl  = blockIdx.x;          // column slice
    const int rA = pair, rB = pair + half;
    const int t  = threadIdx.x;
    const int lane_off = t * VEC;

    const float tmpA = temps[rA], tmpB = temps[rB];
    const bool  sampA = (tmpA != 0.0f), sampB = (tmpB != 0.0f);
    const float invA = sampA ? (1.0f / tmpA) : 1.0f;
    const float invB = sampB ? (1.0f / tmpB) : 1.0f;

    const float* baseA = logits + (size_t)rA * (size_t)V;
    const float* baseB = logits + (size_t)rB * (size_t)V;
    const uint32_t jbase  = (uint32_t)rA * (uint32_t)V;   // flat index of (rA,0)
    const uint32_t jdelta = (uint32_t)half * (uint32_t)V; // N/2 pairing offset

    const int ntiles   = V / TILE;
    const int tilesPer = (ntiles + SPLIT - 1) / SPLIT;
    const int t0 = spl * tilesPer;
    int t1 = t0 + tilesPer; if (t1 > ntiles) t1 = ntiles;
    const int myTiles = (t1 > t0) ? (t1 - t0) : 0;

    // --- pipeline prologue --------------------------------------------------
#pragma unroll
    for (int p = 0; p < STAGES; ++p) {
        if (p < myTiles) {
            async_b128((uint32_t)(uintptr_t)&tiles[p][0][lane_off],
                       baseA + (t0 + p) * TILE + lane_off);
            async_b128((uint32_t)(uintptr_t)&tiles[p][1][lane_off],
                       baseB + (t0 + p) * TILE + lane_off);
        }
    }

    float gVA = -INFINITY, gVB = -INFINITY, sVA = -INFINITY, sVB = -INFINITY;
    int   gIA = 0, gIB = 0, sIA = 0, sIB = 0;

    auto update = [&](float av, float bv, int col, uint32_t o0, uint32_t o1) {
        if (av > gVA) { gVA = av; gIA = col; }
        if (bv > gVB) { gVB = bv; gIB = col; }
        if (sampA) {                                  // wave-uniform branch
            float s = fmaf(av, invA, gumbel_from_bits(o0));
            if (s > sVA) { sVA = s; sIA = col; }
        }
        if (sampB) {
            float s = fmaf(bv, invB, gumbel_from_bits(o1));
            if (s > sVB) { sVB = s; sIB = col; }
        }
    };

    auto consume = [&](int tl) {    // tl = local tile index
        const int buf = tl & (STAGES - 1);
        float4 a4 = *(const float4*)&tiles[buf][0][lane_off]; // ds_load_b128
        float4 b4 = *(const float4*)&tiles[buf][1][lane_off];
        float av[VEC] = {a4.x, a4.y, a4.z, a4.w};
        float bv[VEC] = {b4.x, b4.y, b4.z, b4.w};
        const int colBase = (t0 + tl) * TILE + lane_off;
#pragma unroll
        for (int k = 0; k < VEC; ++k) {
            const int col = colBase + k;
            uint32_t o0, o1;
            threefry2x32_k1(jbase + (uint32_t)col,
                            jbase + (uint32_t)col + jdelta, o0, o1);
            update(av[k], bv[k], col, o0, o1);
        }
    };

    // --- steady state: branch-free wait(6) + unconditional re-issue ---------
    int tl = 0;
    const float* issA = baseA + (t0 + STAGES) * TILE + lane_off;
    const float* issB = baseB + (t0 + STAGES) * TILE + lane_off;
    for (; tl + STAGES < myTiles; ++tl) {
        WAIT_ASYNC(6);
        consume(tl);
        WAIT_DS0();  // WAR guard: this buffer's ds reads retired
        const int buf = tl & (STAGES - 1);
        async_b128((uint32_t)(uintptr_t)&tiles[buf][0][lane_off], issA);
        async_b128((uint32_t)(uintptr_t)&tiles[buf][1][lane_off], issB);
        issA += TILE; issB += TILE;
    }
    // --- epilogue: drain last <=STAGES tiles --------------------------------
    for (; tl < myTiles; ++tl) {
        const int ahead = myTiles - tl - 1;  // <= 3 here
        switch (ahead) {
            case 3:  WAIT_ASYNC(6); break;
            case 2:  WAIT_ASYNC(4); break;
            case 1:  WAIT_ASYNC(2); break;
            default: WAIT_ASYNC(0); break;
        }
        consume(tl);
    }

    // --- tail columns (V % TILE): last slice only, direct loads -------------
    if (spl == SPLIT - 1) {
        for (int col = ntiles * TILE + t; col < V; col += THREADS) {
            uint32_t o0, o1;
            threefry2x32_k1(jbase + (uint32_t)col,
                            jbase + (uint32_t)col + jdelta, o0, o1);
            update(baseA[col], baseB[col], col, o0, o1);
        }
    }

    // --- block reduction, first-index tie-break ------------------------------
    redV[0][t] = gVA; redI[0][t] = gIA;
    redV[1][t] = gVB; redI[1][t] = gIB;
    redV[2][t] = sVA; redI[2][t] = sIA;
    redV[3][t] = sVB; redI[3][t] = sIB;
    __syncthreads();
    for (int s = THREADS >> 1; s > 0; s >>= 1) {
        if (t < s) {
#pragma unroll
            for (int c = 0; c < 4; ++c) {
                float v2 = redV[c][t + s]; int i2 = redI[c][t + s];
                float v1 = redV[c][t];     int i1 = redI[c][t];
                if (v2 > v1 || (v2 == v1 && i2 < i1)) {
                    redV[c][t] = v2; redI[c][t] = i2;
                }
            }
        }
        __syncthreads();
    }
    if (t == 0) {
        const int slot = (pair * SPLIT + spl) * 4;
#pragma unroll
        for (int c = 0; c < 4; ++c) {
            wsV[slot + c] = redV[c][0];
            wsI[slot + c] = redI[c][0];
        }
    }
}

// Second pass: fold SPLIT partials per row pair, apply temp==0 selection.
__global__ __launch_bounds__(128)
void sampler_finalize_kernel(const float* __restrict__ temps,
                             const float* __restrict__ wsV,
                             const int* __restrict__ wsI,
                             int* __restrict__ out, int B) {
    const int half = B >> 1;
    const int pair = blockIdx.x * blockDim.x + threadIdx.x;
    if (pair >= half) return;

    float v[4]  = {-INFINITY, -INFINITY, -INFINITY, -INFINITY};
    int   idx[4] = {0, 0, 0, 0};
    for (int s = 0; s < SPLIT; ++s) {          // ascending column ranges
        const int slot = (pair * SPLIT + s) * 4;
#pragma unroll
        for (int c = 0; c < 4; ++c) {
            float v2 = wsV[slot + c]; int i2 = wsI[slot + c];
            if (v2 > v[c] || (v2 == v[c] && i2 < idx[c])) {
                v[c] = v2; idx[c] = i2;
            }
        }
    }
    const float tA = temps[pair], tB = temps[pair + half];
    out[pair]        = (tA != 0.0f) ? idx[2] : idx[0];
    out[pair + half] = (tB != 0.0f) ? idx[3] : idx[1];
}

extern "C" void kernel_launch(void* const* d_in, const int* in_sizes, int n_in,
                              void* d_out, int out_size, void* d_ws, size_t ws_size,
                              hipStream_t stream) {
    const float* logits = (const float*)d_in[0];
    const float* temps  = (const float*)d_in[1];
    const int B = in_sizes[1];            // 256
    const int V = in_sizes[0] / B;        // 128000
    const int half = B / 2;
    int* out = (int*)d_out;               // int32 tokens

    // workspace: half*SPLIT*4 floats then half*SPLIT*4 ints
    float* wsV = (float*)d_ws;
    int*   wsI = (int*)d_ws + (size_t)half * SPLIT * 4;

    dim3 grid1(SPLIT, half), block1(THREADS);
    sampler_partial_kernel<<<grid1, block1, 0, stream>>>(logits, temps,
                                                         wsV, wsI, B, V);
    dim3 grid2((half + 127) / 128), block2(128);
    sampler_finalize_kernel<<<grid2, block2, 0, stream>>>(temps, wsV, wsI,
                                                          out, B);
}